// SelfCorrectingPointProcess_12412455486045
// MI455X (gfx1250) — compile-verified
//
#include <hip/hip_runtime.h>
#include <hip/hip_bf16.h>
#include <cstdint>

// SelfCorrectingPointProcess: per-row masked scan, bandwidth-bound (128 MB in,
// ~5.5 us floor @ 23.3 TB/s). No matrix ops -> WMMA inapplicable; use the
// CDNA5 Tensor Data Mover (tensor_load_to_lds + s_wait_tensorcnt) to
// double-buffer row tiles into LDS while the wave32 shuffle-scan consumes them.

#define T_LEN 2048
#define CHUNK 256
#define NCHUNK (T_LEN / CHUNK) // 8
#define ROWS_PER_BLOCK 8
#define SUBS (CHUNK / 128)     // 2

typedef unsigned int u32;
typedef __attribute__((ext_vector_type(4))) u32 u32x4_t;
typedef __attribute__((ext_vector_type(8))) int i32x8_t;
typedef __attribute__((ext_vector_type(4))) int i32x4_t;

#if __has_builtin(__builtin_amdgcn_tensor_load_to_lds) && __has_builtin(__builtin_amdgcn_s_wait_tensorcnt)
#define USE_TDM 1
#endif

#ifdef USE_TDM
#define TDM_WAIT(n) __builtin_amdgcn_s_wait_tensorcnt((short)(n))
#else
#define TDM_WAIT(n)
#endif

__device__ __forceinline__ u32 lds_addr_of(const void* p) {
  // LDS-aperture flat pointers carry the LDS byte offset in addr[31:0].
  return (u32)(uintptr_t)p;
}

// Stage CHUNK consecutive floats from global into this wave's LDS buffer.
__device__ __forceinline__ void stage_row(float* ldst, const float* gsrc, int lane) {
#ifdef USE_TDM
  (void)lane;
  unsigned long long ga = (unsigned long long)(uintptr_t)gsrc;
  u32x4_t g0;
  g0[0] = 1u;                                              // count=1 (valid user D#)
  g0[1] = lds_addr_of(ldst);                               // lds_addr (bytes)
  g0[2] = (u32)(ga & 0xFFFFFFFFull);                       // global_addr[31:0]
  g0[3] = (u32)((ga >> 32) & 0x01FFFFFFull) | (2u << 30);  // global_addr[56:32] | type=2
  i32x8_t g1;
  g1[0] = 0x20000;                        // data_size = 2 -> 4 bytes; no mask/pad/iterate
  g1[1] = (int)((CHUNK & 0xFFFF) << 16);  // tensor_dim0[15:0] = CHUNK
  g1[2] = (int)(1u << 16);                // tensor_dim0 hi = 0 | tensor_dim1 = 1
  g1[3] = (int)((u32)CHUNK << 16);        // tensor_dim1 hi = 0 | tile_dim0 = CHUNK
  g1[4] = 1;                              // tile_dim1 = 1, tile_dim2 = 0
  g1[5] = T_LEN;                          // tensor_dim0_stride[31:0]
  g1[6] = 0;                              // stride hi / tensor_dim1_stride lo
  g1[7] = 0;
  i32x4_t z4 = {0, 0, 0, 0};              // groups 2/3 unused (<=2D tensor)
  i32x8_t z8 = {0, 0, 0, 0, 0, 0, 0, 0};
  // 6-arg toolchain variant: (g0, g1, g2, g3, extra, cpol)
  __builtin_amdgcn_tensor_load_to_lds(g0, g1, z4, z4, z8, 0);
#else
  #pragma unroll
  for (int i = 0; i < CHUNK / 128; ++i) {
    float4 v = *(const float4*)(gsrc + i * 128 + lane * 4);
    *(float4*)(ldst + i * 128 + lane * 4) = v;
  }
#endif
}

__global__ __launch_bounds__(256) void scpp_kernel(
    const float* __restrict__ et, const float* __restrict__ im,
    const float* __restrict__ t0p, const float* __restrict__ t1p,
    const float* __restrict__ mup, const float* __restrict__ betap,
    float* __restrict__ out, int nrows)
{
  __shared__ float bT[2][ROWS_PER_BLOCK][CHUNK];
  __shared__ float bM[2][ROWS_PER_BLOCK][CHUNK];

  const int lane = threadIdx.x & 31;
  const int w    = threadIdx.x >> 5;
  const int row  = blockIdx.x * ROWS_PER_BLOCK + w;
  if (row >= nrows) return;

  const float t0v  = t0p[0];
  const float t1v  = t1p[0];
  const float mu   = log1pf(__expf(mup[0]));    // softplus
  const float beta = log1pf(__expf(betap[0]));  // softplus

  const float* rowT = et + (size_t)row * T_LEN;
  const float* rowM = im + (size_t)row * T_LEN;

  // Prologue: stage chunk 0 into buffer 0.
  stage_row(&bT[0][w][0], rowT, lane);
  stage_row(&bM[0][w][0], rowM, lane);

  int   cntBase = 0;      // masked-event count before current chunk (uniform)
  float prevT   = t0v;    // last masked event time so far (uniform)
  float compAcc = 0.f, tAcc = 0.f, idxAcc = 0.f;  // per-lane partials

  for (int c = 0; c < NCHUNK; ++c) {
    const int cb = c & 1;
    if (c + 1 < NCHUNK) {
      const int nb = (c + 1) & 1;
      stage_row(&bT[nb][w][0], rowT + (c + 1) * CHUNK, lane);
      stage_row(&bM[nb][w][0], rowM + (c + 1) * CHUNK, lane);
      TDM_WAIT(2);  // two newest in flight; chunk c's pair has landed
    } else {
      TDM_WAIT(0);
    }

    #pragma unroll
    for (int s = 0; s < SUBS; ++s) {
      const int off = s * 128 + lane * 4;
      float4 tv = *(const float4*)&bT[cb][w][off];
      float4 mv = *(const float4*)&bM[cb][w][off];

      float t[4]    = {tv.x, tv.y, tv.z, tv.w};
      float mval[4] = {mv.x, mv.y, mv.z, mv.w};
      int msk[4]; int cnt_l = 0; int has = 0; float lastT = 0.f;
      #pragma unroll
      for (int k = 0; k < 4; ++k) {
        msk[k] = mval[k] > 0.5f;
        cnt_l += msk[k];
        if (msk[k]) { lastT = t[k]; has = 1; }
      }

      // Wave32 inclusive scans: masked count + forward-fill of last masked t.
      int cs = cnt_l; int h = has; float ft = lastT;
      #pragma unroll
      for (int d = 1; d < 32; d <<= 1) {
        int   cy = __shfl_up(cs, d, 32);
        int   hy = __shfl_up(h, d, 32);
        float fy = __shfl_up(ft, d, 32);
        if (lane >= d) {
          cs += cy;
          if (!h) { h = hy; ft = fy; }
        }
      }
      const int   totalCnt = __shfl(cs, 31, 32);
      const int   hTot     = __shfl(h, 31, 32);
      const float ftTot    = __shfl(ft, 31, 32);
      int   hE  = __shfl_up(h, 1, 32);
      float ftE = __shfl_up(ft, 1, 32);
      if (lane == 0) hE = 0;

      int   e  = cntBase + (cs - cnt_l);        // exclusive masked count at first elem
      float pt = hE ? ftE : prevT;              // previous masked time entering this lane
      const int ebase = c * CHUNK + s * 128 + lane * 4;
      #pragma unroll
      for (int k = 0; k < 4; ++k) {
        if (msk[k]) {
          const float term = __expf(-beta * (float)e) *
                             (__expf(mu * t[k]) - __expf(mu * pt)) / mu;
          compAcc += term;
          tAcc    += t[k];
          idxAcc  += (float)(ebase + k);        // exact: integers < 2^24
          pt = t[k];
          ++e;
        }
      }
      cntBase += totalCnt;
      if (hTot) prevT = ftTot;
    }
  }

  // Wave reduction of partials.
  #pragma unroll
  for (int d = 16; d >= 1; d >>= 1) {
    compAcc += __shfl_xor(compAcc, d, 32);
    tAcc    += __shfl_xor(tAcc, d, 32);
    idxAcc  += __shfl_xor(idxAcc, d, 32);
  }
  if (lane == 0) {
    const float tail = __expf(-beta * (float)cntBase) *
                       (__expf(mu * t1v) - __expf(mu * prevT)) / mu;
    out[row] = (mu * tAcc - beta * idxAcc) - (compAcc + tail);
  }
}

extern "C" void kernel_launch(void* const* d_in, const int* in_sizes, int n_in,
                              void* d_out, int out_size, void* d_ws, size_t ws_size,
                              hipStream_t stream) {
  (void)n_in; (void)out_size; (void)d_ws; (void)ws_size;
  const float* et    = (const float*)d_in[0];  // event_times (N,T)
  // d_in[1] spatial_locations: unused by the reference
  const float* im    = (const float*)d_in[2];  // input_mask (N,T)
  const float* t0p   = (const float*)d_in[3];
  const float* t1p   = (const float*)d_in[4];
  const float* mup   = (const float*)d_in[5];
  const float* betap = (const float*)d_in[6];
  float* out = (float*)d_out;

  const int nrows  = in_sizes[0] / T_LEN;
  const int blocks = (nrows + ROWS_PER_BLOCK - 1) / ROWS_PER_BLOCK;
  hipLaunchKernelGGL(scpp_kernel, dim3(blocks), dim3(256), 0, stream,
                     et, im, t0p, t1p, mup, betap, out, nrows);
}